// POELayer_36928128811077
// MI455X (gfx1250) — compile-verified
//
#include <hip/hip_runtime.h>
#include <math.h>

typedef __attribute__((ext_vector_type(2))) float vf2;
typedef __attribute__((ext_vector_type(4))) float vf4;

#define NJ        12
#define EPSF      1e-10f
#define JSTRIDE   32            // floats per joint constant block
#define TB_OFF    (NJ * JSTRIDE)
#define TB_STRIDE 16            // floats per branch transform block
#define BLOCK     256
#define OUT_LSTR  68            // padded LDS stride for 64-float outs rows (2-way banks)
#define TW_LSTR   76            // padded LDS stride for 72-float twl rows  (2-way banks)

#ifndef __has_builtin
#define __has_builtin(x) 0
#endif
#if __has_builtin(__builtin_amdgcn_global_store_async_from_lds_b128)
#define HAVE_ASYNC_LDS 1
#define AS_GLOBAL __attribute__((address_space(1)))
#define AS_LOCAL  __attribute__((address_space(3)))
typedef int vsi4 __attribute__((vector_size(4 * sizeof(int))));   // matches builtin param type
#else
#define HAVE_ASYNC_LDS 0
#endif

// ---------------------------------------------------------------------------
// Kernel 1: fold all joint-invariant math into a small constant table in d_ws.
// Layout per joint j (cb + j*32):
//   [0] wn  [1..3] wh  [4..6] vh  [7..9] wh x vh  [10..12] K2*vh
//   [13..15] K2 diag (wh_i^2 - |wh|^2)  [16..18] K2 off-diag (xy, xz, yz)
//   [19..21] w  [22..24] v
// Branch transforms (top 3x4 rows, row-major) at cb + TB_OFF + j*16.
// ---------------------------------------------------------------------------
__global__ void poe_precompute(const float* __restrict__ twist,
                               const float* __restrict__ branch_p,
                               const float* __restrict__ branch_rpy,
                               float* __restrict__ cb)
{
    int j = threadIdx.x;
    if (j >= NJ) return;

    const float* t = twist + j * 6;
    float wx = t[0], wy = t[1], wz = t[2];
    float vx = t[3], vy = t[4], vz = t[5];

    float wn  = sqrtf(wx * wx + wy * wy + wz * wz);
    float inv = 1.0f / (wn + EPSF);
    float whx = wx * inv, why = wy * inv, whz = wz * inv;
    float ax  = vx * inv, ay  = vy * inv, az  = vz * inv;          // vh
    float bx  = why * az - whz * ay;                               // wh x vh
    float by  = whz * ax - whx * az;
    float bz  = whx * ay - why * ax;
    float n2  = whx * whx + why * why + whz * whz;                 // |wh|^2
    float wdv = whx * ax + why * ay + whz * az;                    // wh . vh
    float dx  = whx * wdv - n2 * ax;                               // K2 * vh
    float dy  = why * wdv - n2 * ay;
    float dz  = whz * wdv - n2 * az;

    float* c = cb + j * JSTRIDE;
    c[0]  = wn;
    c[1]  = whx; c[2]  = why; c[3]  = whz;
    c[4]  = ax;  c[5]  = ay;  c[6]  = az;
    c[7]  = bx;  c[8]  = by;  c[9]  = bz;
    c[10] = dx;  c[11] = dy;  c[12] = dz;
    c[13] = whx * whx - n2;                                        // K2[0][0]
    c[14] = why * why - n2;                                        // K2[1][1]
    c[15] = whz * whz - n2;                                        // K2[2][2]
    c[16] = whx * why;                                             // K2[0][1]
    c[17] = whx * whz;                                             // K2[0][2]
    c[18] = why * whz;                                             // K2[1][2]
    c[19] = wx;  c[20] = wy;  c[21] = wz;
    c[22] = vx;  c[23] = vy;  c[24] = vz;

    // Branch transform: R = Rz(y) * Ry(pitch) * Rx(roll), p = branch_p[j]
    float r  = branch_rpy[j * 3 + 0];
    float pc = branch_rpy[j * 3 + 1];
    float yv = branch_rpy[j * 3 + 2];
    float cr = cosf(r),  sr = sinf(r);
    float cp = cosf(pc), sp = sinf(pc);
    float cy = cosf(yv), sy = sinf(yv);

    float* tb = cb + TB_OFF + j * TB_STRIDE;
    tb[0]  = cy * cp;  tb[1]  = cy * sp * sr - sy * cr;  tb[2]  = cy * sp * cr + sy * sr;  tb[3]  = branch_p[j * 3 + 0];
    tb[4]  = sy * cp;  tb[5]  = sy * sp * sr + cy * cr;  tb[6]  = sy * sp * cr - cy * sr;  tb[7]  = branch_p[j * 3 + 1];
    tb[8]  = -sp;      tb[9]  = cp * sr;                 tb[10] = cp * cr;                 tb[11] = branch_p[j * 3 + 2];
}

// ---------------------------------------------------------------------------
// Kernel 2: one thread per batch element; running transform in 12 registers.
// Results staged in bank-padded LDS rows, then drained with perfectly
// coalesced b128 stores via the gfx1250 async LDS->global DMA
// (global_store_async_from_lds_b128, ASYNCcnt), with a ds_load + NT-store
// fallback. S_ENDPGM's implicit wait-idle drains ASYNCcnt before LDS frees.
// ---------------------------------------------------------------------------
__global__ __launch_bounds__(BLOCK) void poe_main(
    const float* __restrict__ q,
    const float* __restrict__ cb,
    float* __restrict__ outs,
    float* __restrict__ twl,
    int B)
{
    __shared__ __align__(16) float s_outs[BLOCK * OUT_LSTR];   // 68 KB
    __shared__ __align__(16) float s_twl [BLOCK * TW_LSTR];    // 76 KB

    const int tid       = threadIdx.x;
    const int blockBase = blockIdx.x * BLOCK;
    const int idx       = blockBase + tid;

    if (idx < B) {
        // 12 q values per element: three b128 non-temporal loads
        const float* qp = q + (size_t)idx * 12;
        vf4 qa = __builtin_nontemporal_load((const vf4*)qp);
        vf4 qb = __builtin_nontemporal_load((const vf4*)(qp + 4));
        vf4 qc = __builtin_nontemporal_load((const vf4*)(qp + 8));
        float qv[12] = {qa.x, qa.y, qa.z, qa.w,
                        qb.x, qb.y, qb.z, qb.w,
                        qc.x, qc.y, qc.z, qc.w};

        // Running transform (top 3 rows of 4x4), starts as identity.
        float t00 = 1.f, t01 = 0.f, t02 = 0.f, t03 = 0.f;
        float t10 = 0.f, t11 = 1.f, t12 = 0.f, t13 = 0.f;
        float t20 = 0.f, t21 = 0.f, t22 = 1.f, t23 = 0.f;

        float* ltw = s_twl  + tid * TW_LSTR;
        float* lot = s_outs + tid * OUT_LSTR;

#pragma unroll
        for (int j = 0; j < NJ; ++j) {
            const float* c = cb + j * JSTRIDE;   // uniform address -> scalar loads
            float wn  = c[0];
            float whx = c[1],  why = c[2],  whz = c[3];
            float ax  = c[4],  ay  = c[5],  az  = c[6];
            float bx  = c[7],  by  = c[8],  bz  = c[9];
            float dx  = c[10], dy  = c[11], dz  = c[12];
            float dxx = c[13], dyy = c[14], dzz = c[15];
            float kxy = c[16], kxz = c[17], kyz = c[18];
            float wx  = c[19], wy  = c[20], wz  = c[21];
            float vx  = c[22], vy  = c[23], vz  = c[24];

            // ---- twist_ls[j] = AdInv(T) * twist[j]  (uses T *before* this joint)
            float wox = t00 * wx + t10 * wy + t20 * wz;       // R^T w
            float woy = t01 * wx + t11 * wy + t21 * wz;
            float woz = t02 * wx + t12 * wy + t22 * wz;
            float pxwx = t13 * wz - t23 * wy;                 // p x w, p = T[:,3]
            float pxwy = t23 * wx - t03 * wz;
            float pxwz = t03 * wy - t13 * wx;
            float ux = vx - pxwx, uy = vy - pxwy, uz = vz - pxwz;
            float vox = t00 * ux + t10 * uy + t20 * uz;       // R^T (v - p x w)
            float voy = t01 * ux + t11 * uy + t21 * uz;
            float voz = t02 * ux + t12 * uy + t22 * uz;

            {   // 6 floats, 8B aligned -> ds b64 stores
                float* o = ltw + j * 6;
                vf2 s0 = {wox, woy};
                vf2 s1 = {woz, vox};
                vf2 s2 = {voy, voz};
                *(vf2*)(o)     = s0;
                *(vf2*)(o + 2) = s1;
                *(vf2*)(o + 4) = s2;
            }

            // ---- E = screw_exp(twist[j], q_j):  R = I + s K + (1-c) K^2,
            //      p = th*vh + (1-c)*(K vh) + (th - s)*(K^2 vh)
            float th = qv[j] * wn;
            float s, cth;
            sincosf(th, &s, &cth);
            float omc = 1.0f - cth;
            float tms = th - s;

            float e00 = 1.0f + omc * dxx;
            float e01 = omc * kxy - s * whz;
            float e02 = omc * kxz + s * why;
            float e10 = omc * kxy + s * whz;
            float e11 = 1.0f + omc * dyy;
            float e12 = omc * kyz - s * whx;
            float e20 = omc * kxz - s * why;
            float e21 = omc * kyz + s * whx;
            float e22 = 1.0f + omc * dzz;
            float px  = th * ax + omc * bx + tms * dx;
            float py  = th * ay + omc * by + tms * dy;
            float pz  = th * az + omc * bz + tms * dz;

            // ---- T = T * E   (E bottom row [0,0,0,1])
            float n00 = t00 * e00 + t01 * e10 + t02 * e20;
            float n01 = t00 * e01 + t01 * e11 + t02 * e21;
            float n02 = t00 * e02 + t01 * e12 + t02 * e22;
            float n03 = t00 * px  + t01 * py  + t02 * pz + t03;
            float n10 = t10 * e00 + t11 * e10 + t12 * e20;
            float n11 = t10 * e01 + t11 * e11 + t12 * e21;
            float n12 = t10 * e02 + t11 * e12 + t12 * e22;
            float n13 = t10 * px  + t11 * py  + t12 * pz + t13;
            float n20 = t20 * e00 + t21 * e10 + t22 * e20;
            float n21 = t20 * e01 + t21 * e11 + t22 * e21;
            float n22 = t20 * e02 + t21 * e12 + t22 * e22;
            float n23 = t20 * px  + t21 * py  + t22 * pz + t23;
            t00 = n00; t01 = n01; t02 = n02; t03 = n03;
            t10 = n10; t11 = n11; t12 = n12; t13 = n13;
            t20 = n20; t21 = n21; t22 = n22; t23 = n23;

            // ---- branch output at joints 2, 5, 8, 11
            if (j == 2 || j == 5 || j == 8 || j == 11) {
                const int ob = j / 3;                      // 0..3
                const float* tb = cb + TB_OFF + j * TB_STRIDE;
                float o00 = t00 * tb[0] + t01 * tb[4] + t02 * tb[8];
                float o01 = t00 * tb[1] + t01 * tb[5] + t02 * tb[9];
                float o02 = t00 * tb[2] + t01 * tb[6] + t02 * tb[10];
                float o03 = t00 * tb[3] + t01 * tb[7] + t02 * tb[11] + t03;
                float o10 = t10 * tb[0] + t11 * tb[4] + t12 * tb[8];
                float o11 = t10 * tb[1] + t11 * tb[5] + t12 * tb[9];
                float o12 = t10 * tb[2] + t11 * tb[6] + t12 * tb[10];
                float o13 = t10 * tb[3] + t11 * tb[7] + t12 * tb[11] + t13;
                float o20 = t20 * tb[0] + t21 * tb[4] + t22 * tb[8];
                float o21 = t20 * tb[1] + t21 * tb[5] + t22 * tb[9];
                float o22 = t20 * tb[2] + t21 * tb[6] + t22 * tb[10];
                float o23 = t20 * tb[3] + t21 * tb[7] + t22 * tb[11] + t23;

                float* o = lot + ob * 16;
                vf4 r0 = {o00, o01, o02, o03};
                vf4 r1 = {o10, o11, o12, o13};
                vf4 r2 = {o20, o21, o22, o23};
                vf4 r3 = {0.f, 0.f, 0.f, 1.f};
                *(vf4*)(o)      = r0;
                *(vf4*)(o + 4)  = r1;
                *(vf4*)(o + 8)  = r2;
                *(vf4*)(o + 12) = r3;
            }
        }
    }

    __syncthreads();   // DS writes visible before the coalesced drain

    int nvalid = B - blockBase;
    if (nvalid > BLOCK) nvalid = BLOCK;

    float* gouts = outs + (size_t)blockBase * 64;
    float* gtwl  = twl  + (size_t)blockBase * 72;

#if HAVE_ASYNC_LDS
    // gfx1250 async DMA drain: lane L of chunk c moves 16B, fully coalesced
    // (512B contiguous per wave per instruction). ASYNCcnt drains at S_ENDPGM.
#pragma unroll
    for (int c = 0; c < 16; ++c) {
        int jdx = c * BLOCK + tid;
        if (jdx < nvalid * 16) {
            int e = jdx >> 4, k = jdx & 15;
            __builtin_amdgcn_global_store_async_from_lds_b128(
                (AS_GLOBAL vsi4*)(gouts + (size_t)jdx * 4),
                (AS_LOCAL  vsi4*)(s_outs + e * OUT_LSTR + k * 4), 0, 0);
        }
    }
#pragma unroll
    for (int c = 0; c < 18; ++c) {
        int jdx = c * BLOCK + tid;
        if (jdx < nvalid * 18) {
            int e = jdx / 18, k = jdx - e * 18;
            __builtin_amdgcn_global_store_async_from_lds_b128(
                (AS_GLOBAL vsi4*)(gtwl + (size_t)jdx * 4),
                (AS_LOCAL  vsi4*)(s_twl + e * TW_LSTR + k * 4), 0, 0);
        }
    }
#else
    // Fallback: ds_load_b128 + coalesced nontemporal global_store_b128
#pragma unroll
    for (int c = 0; c < 16; ++c) {
        int jdx = c * BLOCK + tid;
        if (jdx < nvalid * 16) {
            int e = jdx >> 4, k = jdx & 15;
            vf4 v = *(const vf4*)(s_outs + e * OUT_LSTR + k * 4);
            __builtin_nontemporal_store(v, (vf4*)(gouts + (size_t)jdx * 4));
        }
    }
#pragma unroll
    for (int c = 0; c < 18; ++c) {
        int jdx = c * BLOCK + tid;
        if (jdx < nvalid * 18) {
            int e = jdx / 18, k = jdx - e * 18;
            vf4 v = *(const vf4*)(s_twl + e * TW_LSTR + k * 4);
            __builtin_nontemporal_store(v, (vf4*)(gtwl + (size_t)jdx * 4));
        }
    }
#endif
}

extern "C" void kernel_launch(void* const* d_in, const int* in_sizes, int n_in,
                              void* d_out, int out_size, void* d_ws, size_t ws_size,
                              hipStream_t stream)
{
    const float* q_value    = (const float*)d_in[0];   // (B, 12)
    const float* twist      = (const float*)d_in[1];   // (12, 6)
    const float* branch_p   = (const float*)d_in[2];   // (12, 3)
    const float* branch_rpy = (const float*)d_in[3];   // (12, 3)

    const int B = in_sizes[0] / NJ;

    float* cb   = (float*)d_ws;                 // 576 floats of joint constants
    float* outs = (float*)d_out;                // (B, 4, 4, 4)
    float* twl  = outs + (size_t)B * 64;        // (B, 12, 6)

    poe_precompute<<<1, 32, 0, stream>>>(twist, branch_p, branch_rpy, cb);
    poe_main<<<(B + BLOCK - 1) / BLOCK, BLOCK, 0, stream>>>(q_value, cb, outs, twl, B);
}